// GNNNet_38620345925784
// MI455X (gfx1250) — compile-verified
//
#include <hip/hip_runtime.h>

typedef float v2f __attribute__((ext_vector_type(2)));
typedef float v8f __attribute__((ext_vector_type(8)));

#define N_NODES 100000

__device__ __forceinline__ float prelu_f(float v, float a) {
    return v > 0.0f ? v : a * v;
}

// gfx1250 async global->LDS copy, 4 bytes per lane (per-lane LDS dst + global src).
// Tracked by ASYNCcnt; fence with s_wait_asynccnt before reading LDS.
__device__ __forceinline__ void async_copy_b32(unsigned lds_byte_off, const void* gptr) {
    asm volatile("global_load_async_to_lds_b32 %0, %1, off"
                 :: "v"(lds_byte_off), "v"(gptr)
                 : "memory");
}
__device__ __forceinline__ void async_wait0() {
    asm volatile("s_wait_asynccnt 0x0" ::: "memory");
}

// ---------------- degree / normalization ----------------
__global__ void k_deg_init(float* __restrict__ deg, int n) {
    int i = blockIdx.x * blockDim.x + threadIdx.x;
    if (i < n) deg[i] = 1.0f;                 // self-loop weight
}

__global__ void k_deg_accum(float* __restrict__ deg, const long long* __restrict__ ei,
                            const float* __restrict__ w, int E) {
    int e = blockIdx.x * blockDim.x + threadIdx.x;
    if (e < E) {
        long long c = ei[(long long)E + e];
        atomicAdd(&deg[c], w[e]);
    }
}

__global__ void k_deg_inv(float* __restrict__ deg, int n) {
    int i = blockIdx.x * blockDim.x + threadIdx.x;
    if (i < n) {
        float d = deg[i];
        deg[i] = d > 0.0f ? rsqrtf(d) : 0.0f;
    }
}

// ---------------- fp32 WMMA GEMM:  Y[M,64] = X[M,K] @ W[K,64] ----------------
// One wave computes a 16-row x 64-col tile using V_WMMA_F32_16X16X4_F32.
// A 16x4 f32 frag: lane&15 = M row; lanes 0-15 hold K+0/K+1, lanes 16-31 K+2/K+3.
// B 4x16 f32 frag: lane&15 = N col; same K split across lane halves.
// W staged TRANSPOSED in LDS (Wt[n*(KC+2) + k]) so each B-frag is one ds_load_b64
// into an even-aligned VGPR pair; pad +2 keeps 16 reader lanes on distinct banks.
template<int K, int KC>
__global__ __launch_bounds__(256) void k_gemm_wmma(const float* __restrict__ X,
                                                   const float* __restrict__ W,
                                                   float* __restrict__ Y, int M) {
    constexpr int KCP = KC + 2;
    __shared__ float Wt[64 * KCP];
    const int lane = threadIdx.x & 31;
    const int wave = threadIdx.x >> 5;
    const int rowBase = (blockIdx.x * 8 + wave) * 16;
    int r = rowBase + (lane & 15);
    if (r >= M) r = M - 1;                    // clamp: keep EXEC all-ones for WMMA
    const float* __restrict__ xrow = X + (long long)r * K;
    const int koff = (lane >> 4) << 1;        // 0 or 2
    const int col  = lane & 15;
    v8f acc0 = {}, acc1 = {}, acc2 = {}, acc3 = {};
    for (int k0 = 0; k0 < K; k0 += KC) {
        __syncthreads();
        for (int i = threadIdx.x; i < KC * 64; i += 256) {
            int k = i >> 6, n = i & 63;       // W row k0+k, col n
            async_copy_b32((unsigned)(size_t)&Wt[n * KCP + k], W + k0 * 64 + i);
        }
        async_wait0();
        __syncthreads();
        if (k0 + KC < K) __builtin_prefetch(xrow + k0 + KC, 0, 0);
        #pragma unroll 4
        for (int kb = 0; kb < KC; kb += 4) {
            const int kk = kb + koff;
            v2f a  = *(const v2f*)(xrow + k0 + kk);
            v2f b0 = *(const v2f*)&Wt[(col +  0) * KCP + kk];
            v2f b1 = *(const v2f*)&Wt[(col + 16) * KCP + kk];
            v2f b2 = *(const v2f*)&Wt[(col + 32) * KCP + kk];
            v2f b3 = *(const v2f*)&Wt[(col + 48) * KCP + kk];
            acc0 = __builtin_amdgcn_wmma_f32_16x16x4_f32(false, a, false, b0, (short)0, acc0, false, false);
            acc1 = __builtin_amdgcn_wmma_f32_16x16x4_f32(false, a, false, b1, (short)0, acc1, false, false);
            acc2 = __builtin_amdgcn_wmma_f32_16x16x4_f32(false, a, false, b2, (short)0, acc2, false, false);
            acc3 = __builtin_amdgcn_wmma_f32_16x16x4_f32(false, a, false, b3, (short)0, acc3, false, false);
        }
    }
    const int mBase = rowBase + ((lane >> 4) << 3);
    #pragma unroll
    for (int i = 0; i < 8; ++i) {
        int rr = mBase + i;
        if (rr < M) {
            float* y = Y + (long long)rr * 64 + col;
            y[0]  = acc0[i];
            y[16] = acc1[i];
            y[32] = acc2[i];
            y[48] = acc3[i];
        }
    }
}

// ---------------- aggregation (self-loop init + edge scatter) ----------------
__global__ void k_selfloop(float* __restrict__ agg, const float* __restrict__ h,
                           const float* __restrict__ dinv, int total) {
    int i = blockIdx.x * blockDim.x + threadIdx.x;
    if (i < total) {
        float dv = dinv[i >> 6];
        agg[i] = dv * dv * h[i];              // norm = dinv[n]*1*dinv[n]
    }
}

__global__ __launch_bounds__(256) void k_edge_scatter(const float* __restrict__ h,
                                                      const long long* __restrict__ ei,
                                                      const float* __restrict__ w,
                                                      const float* __restrict__ dinv,
                                                      float* __restrict__ agg, int E) {
    const int lane = threadIdx.x & 31;
    int wv = (blockIdx.x * blockDim.x + threadIdx.x) >> 5;
    const int nwv = (gridDim.x * blockDim.x) >> 5;
    for (int e = wv; e < E; e += nwv) {
        long long r = ei[e];
        long long c = ei[(long long)E + e];
        float norm = dinv[r] * w[e] * dinv[c];
        const float2 v = ((const float2*)(h + r * 64))[lane];
        float* dst = agg + c * 64 + lane * 2;
        atomicAdd(dst,     norm * v.x);
        atomicAdd(dst + 1, norm * v.y);
    }
}

__global__ void k_bias_prelu(float* __restrict__ buf, const float* __restrict__ b,
                             const float* __restrict__ alpha, int total) {
    int i = blockIdx.x * blockDim.x + threadIdx.x;
    if (i < total) {
        float v = buf[i] + b[i & 63];
        buf[i] = prelu_f(v, alpha[0]);
    }
}

// ---------------- MLP head epilogue: fused PReLU + W2 dot + butterfly reduce ----------------
__device__ __forceinline__ void head_reduce_store(const v8f& a0, const v8f& a1,
                                                  const v8f& a2, const v8f& a3,
                                                  float b1v0, float b1v1, float b1v2, float b1v3,
                                                  float w2v0, float w2v1, float w2v2, float w2v3,
                                                  float pa, float bo, int lane, int col,
                                                  float* __restrict__ outBase) {
    #pragma unroll
    for (int i = 0; i < 8; ++i) {
        float p = prelu_f(a0[i] + b1v0, pa) * w2v0
                + prelu_f(a1[i] + b1v1, pa) * w2v1
                + prelu_f(a2[i] + b1v2, pa) * w2v2
                + prelu_f(a3[i] + b1v3, pa) * w2v3;
        p += __shfl_xor(p, 1, 32);
        p += __shfl_xor(p, 2, 32);
        p += __shfl_xor(p, 4, 32);
        p += __shfl_xor(p, 8, 32);
        if (col == i)
            outBase[i + ((lane >> 4) << 3)] = p + bo;
    }
}

// ---------------- MLP head: out[l] = W2 . prelu(W1^T [h[a];h[b]] + b1) + b2 ----------------
// 2 M-tiles (32 label edges) per wave: 8 WMMA per {4 B ds_load_b64 + 2 A global b64}.
#define KD  128
#define KDP 130
__global__ __launch_bounds__(256) void k_dnn_wmma(const float* __restrict__ hf,
                                                  const long long* __restrict__ le,
                                                  const float* __restrict__ W1,
                                                  const float* __restrict__ b1,
                                                  const float* __restrict__ pa_p,
                                                  const float* __restrict__ W2,
                                                  const float* __restrict__ b2_p,
                                                  float* __restrict__ out, int L) {
    __shared__ float W1t[64 * KDP];           // transposed + padded, ~33 KB
    for (int i = threadIdx.x; i < KD * 64; i += 256) {
        int k = i >> 6, n = i & 63;
        async_copy_b32((unsigned)(size_t)&W1t[n * KDP + k], W1 + i);
    }
    async_wait0();
    __syncthreads();
    const int lane = threadIdx.x & 31;
    const int wave = threadIdx.x >> 5;
    const int col  = lane & 15;
    const int koff = (lane >> 4) << 1;
    const float pa = pa_p[0];
    const float bo = b2_p[0];
    const float b1v0 = b1[col], b1v1 = b1[col + 16], b1v2 = b1[col + 32], b1v3 = b1[col + 48];
    const float w2v0 = W2[col], w2v1 = W2[col + 16], w2v2 = W2[col + 32], w2v3 = W2[col + 48];
    const int nTiles = L >> 5;                // 32 rows per wave-tile
    for (int t = blockIdx.x * 8 + wave; t < nTiles; t += gridDim.x * 8) {
        const int row0 = (t << 5) + col;
        const int row1 = row0 + 16;
        const long long ia0 = le[row0], ib0 = le[(long long)L + row0];
        const long long ia1 = le[row1], ib1 = le[(long long)L + row1];
        const float* __restrict__ za0 = hf + ia0 * 64;       // z[0:64)
        const float* __restrict__ zb0 = hf + ib0 * 64 - 64;  // z[64:128), offset by kk
        const float* __restrict__ za1 = hf + ia1 * 64;
        const float* __restrict__ zb1 = hf + ib1 * 64 - 64;
        v8f acc00 = {}, acc01 = {}, acc02 = {}, acc03 = {};
        v8f acc10 = {}, acc11 = {}, acc12 = {}, acc13 = {};
        #pragma unroll 4
        for (int kb = 0; kb < KD; kb += 4) {
            const int kk = kb + koff;
            const bool lo = kk < 64;          // uniform across the wave
            v2f a0 = *(const v2f*)((lo ? za0 : zb0) + kk);
            v2f a1 = *(const v2f*)((lo ? za1 : zb1) + kk);
            v2f b0 = *(const v2f*)&W1t[(col +  0) * KDP + kk];
            v2f b1f = *(const v2f*)&W1t[(col + 16) * KDP + kk];
            v2f b2f = *(const v2f*)&W1t[(col + 32) * KDP + kk];
            v2f b3f = *(const v2f*)&W1t[(col + 48) * KDP + kk];
            acc00 = __builtin_amdgcn_wmma_f32_16x16x4_f32(false, a0, false, b0,  (short)0, acc00, false, false);
            acc01 = __builtin_amdgcn_wmma_f32_16x16x4_f32(false, a0, false, b1f, (short)0, acc01, false, false);
            acc02 = __builtin_amdgcn_wmma_f32_16x16x4_f32(false, a0, false, b2f, (short)0, acc02, false, false);
            acc03 = __builtin_amdgcn_wmma_f32_16x16x4_f32(false, a0, false, b3f, (short)0, acc03, false, false);
            acc10 = __builtin_amdgcn_wmma_f32_16x16x4_f32(false, a1, false, b0,  (short)0, acc10, false, false);
            acc11 = __builtin_amdgcn_wmma_f32_16x16x4_f32(false, a1, false, b1f, (short)0, acc11, false, false);
            acc12 = __builtin_amdgcn_wmma_f32_16x16x4_f32(false, a1, false, b2f, (short)0, acc12, false, false);
            acc13 = __builtin_amdgcn_wmma_f32_16x16x4_f32(false, a1, false, b3f, (short)0, acc13, false, false);
        }
        head_reduce_store(acc00, acc01, acc02, acc03, b1v0, b1v1, b1v2, b1v3,
                          w2v0, w2v1, w2v2, w2v3, pa, bo, lane, col, out + (t << 5));
        head_reduce_store(acc10, acc11, acc12, acc13, b1v0, b1v1, b1v2, b1v3,
                          w2v0, w2v1, w2v2, w2v3, pa, bo, lane, col, out + (t << 5) + 16);
    }
}

// ---------------- host launch ----------------
extern "C" void kernel_launch(void* const* d_in, const int* in_sizes, int n_in,
                              void* d_out, int out_size, void* d_ws, size_t ws_size,
                              hipStream_t stream) {
    const float*     x   = (const float*)d_in[0];
    const long long* ei  = (const long long*)d_in[1];
    const float*     ew  = (const float*)d_in[2];
    const long long* le  = (const long long*)d_in[3];
    const float*     c1W = (const float*)d_in[4];
    const float*     c1b = (const float*)d_in[5];
    const float*     c2W = (const float*)d_in[6];
    const float*     c2b = (const float*)d_in[7];
    const float*     a1  = (const float*)d_in[8];
    const float*     a2  = (const float*)d_in[9];
    const float*     dW1 = (const float*)d_in[10];
    const float*     db1 = (const float*)d_in[11];
    const float*     dpa = (const float*)d_in[12];
    const float*     dW2 = (const float*)d_in[13];
    const float*     db2 = (const float*)d_in[14];
    float* out = (float*)d_out;

    const int N  = N_NODES;
    const int E  = in_sizes[1] / 2;
    const int L  = in_sizes[3] / 2;
    const int NF = N * 64;

    float* bufA = (float*)d_ws;               // 25.6 MB
    float* bufB = bufA + (size_t)NF;          // 25.6 MB
    float* dinv = bufB + (size_t)NF;          // 0.4 MB (deg -> dinv, in place)

    // normalization coefficients
    k_deg_init <<<(N + 255) / 256, 256, 0, stream>>>(dinv, N);
    k_deg_accum<<<(E + 255) / 256, 256, 0, stream>>>(dinv, ei, ew, E);
    k_deg_inv  <<<(N + 255) / 256, 256, 0, stream>>>(dinv, N);

    const int gemmBlocks = (N + 127) / 128;   // 128 rows / block (8 waves x 16 rows)

    // conv1: project (WMMA), self-loop init, scatter, bias+PReLU
    k_gemm_wmma<400, 80><<<gemmBlocks, 256, 0, stream>>>(x, c1W, bufA, N);
    k_selfloop    <<<(NF + 255) / 256, 256, 0, stream>>>(bufB, bufA, dinv, NF);
    k_edge_scatter<<<4096, 256, 0, stream>>>(bufA, ei, ew, dinv, bufB, E);
    k_bias_prelu  <<<(NF + 255) / 256, 256, 0, stream>>>(bufB, c1b, a1, NF);

    // conv2
    k_gemm_wmma<64, 64><<<gemmBlocks, 256, 0, stream>>>(bufB, c2W, bufA, N);
    k_selfloop    <<<(NF + 255) / 256, 256, 0, stream>>>(bufB, bufA, dinv, NF);
    k_edge_scatter<<<4096, 256, 0, stream>>>(bufA, ei, ew, dinv, bufB, E);
    k_bias_prelu  <<<(NF + 255) / 256, 256, 0, stream>>>(bufB, c2b, a2, NF);

    // MLP head over 1M label edges (31250 tiles of 32)
    const int nTiles = L / 32;
    const int dnnBlocks = (nTiles + 7) / 8;
    k_dnn_wmma<<<dnnBlocks, 256, 0, stream>>>(bufB, le, dW1, db1, dpa, dW2, db2, out, L);
}